// DeformableGroupConv_79723182948496
// MI455X (gfx1250) — compile-verified
//
#include <hip/hip_runtime.h>
#include <math.h>

// ---------------------------------------------------------------------------
// Deformable group conv, CDNA5 (gfx1250), wave32.
//   fp32 WMMA (V_WMMA_F32_16X16X4_F32) + async LDS staging + ds_load_b64 paths
//   B=4, C=256, H=W=64, G=8, KH=KW=3, PAD=1, KK=9, CG=32, OffC=2*KK*G=144
// ---------------------------------------------------------------------------

#define BB   4
#define CC   256
#define HH   64
#define WW   64
#define GG   8
#define KKK  9
#define CGC  32
#define OFFC 144          // 2*KK*G
#define HW   (HH * WW)    // 4096
#define KTOT (CGC * KKK)  // 288 (K of the group GEMM)
#define AWS  290          // padded LDS row stride for weight panel (bank spread)
#define S2S  96           // word stride per K-pair row of sampled panel

typedef __attribute__((ext_vector_type(2))) float v2f;
typedef __attribute__((ext_vector_type(8))) float v8f;

// ---------------------------------------------------------------------------
// Kernel R: repack offset_w [144][256][3][3] -> wT [tap][co][ci] so that the
// WMMA A-fragment (K = ci contiguous) is a single 8-byte load.
// ---------------------------------------------------------------------------
__global__ __launch_bounds__(256) void repack_offw(
    const float* __restrict__ offw, float* __restrict__ wT)
{
    const int i = blockIdx.x * 256 + threadIdx.x;     // flat over 144*256*9
    if (i >= OFFC * CC * KKK) return;
    const int tap = i % KKK;
    const int rem = i / KKK;
    const int ci  = rem % CC;
    const int co  = rem / CC;
    wT[((size_t)tap * OFFC + co) * CC + ci] = offw[i];
}

// ---------------------------------------------------------------------------
// Kernel A: offset field = conv3x3(x, offset_w) + offset_b      (implicit GEMM)
// Grid: B * 9(m-tiles of 16 out-ch) * H rows;  block = 128 (4 waves).
// K ordered (tap, ci): per tap, 64 WMMA steps of 4 input channels.
// ---------------------------------------------------------------------------
__global__ __launch_bounds__(128) void offset_conv_wmma(
    const float* __restrict__ x,        // [B][C][H][W]
    const float* __restrict__ wT,       // [9][144][256]   (repacked weights)
    const float* __restrict__ offb,     // [144]
    float* __restrict__ offout)         // [B][144][H][W]  (workspace)
{
    const int bid  = blockIdx.x;
    const int h    = bid % HH;
    const int mt   = (bid / HH) % 9;
    const int b    = bid / (HH * 9);

    const int lane = threadIdx.x & 31;
    const int wv   = threadIdx.x >> 5;
    const int row  = lane & 15;             // M row / N col within tile
    const int kh2  = (lane < 16) ? 0 : 2;   // K sub-offset per WMMA layout
    const int w0   = wv * 16;               // pixel (column) base of this wave

    v8f acc = {};

    for (int tap = 0; tap < 9; ++tap) {
        const int dh  = tap / 3 - 1;
        const int dw  = tap % 3 - 1;
        const int hr  = h + dh;
        const int col = w0 + row + dw;
        const bool vb = (hr >= 0) & (hr < HH) & (col >= 0) & (col < WW);

        // A row: wT[tap][co][ci], contiguous in ci -> one b64 load per step
        const float* wrow = wT + ((size_t)tap * OFFC + (mt * 16 + row)) * CC;
        // B base: x[b][ci][hr][col]
        const float* xbase = x + ((size_t)b * CC) * HW + hr * WW + col;

        for (int ci0 = 0; ci0 < CC; ci0 += 4) {
            const int kx = ci0 + kh2;
            v2f a  = *(const v2f*)&wrow[kx];
            v2f bm;
            bm.x = vb ? xbase[(size_t)kx * HW]       : 0.0f;
            bm.y = vb ? xbase[(size_t)(kx + 1) * HW] : 0.0f;
            acc = __builtin_amdgcn_wmma_f32_16x16x4_f32(
                false, a, false, bm, (short)0, acc, false, false);
        }
    }

    // C/D layout: VGPR v -> M=v (lanes 0-15), M=v+8 (lanes 16-31)
    const int colw = w0 + row;
    const int mh   = (lane < 16) ? 0 : 8;
#pragma unroll
    for (int v = 0; v < 8; ++v) {
        const int co = mt * 16 + v + mh;
        offout[((size_t)b * OFFC + co) * HW + h * WW + colw] = acc[v] + offb[co];
    }
}

// ---------------------------------------------------------------------------
// Kernel B: fused bilinear sampling + per-group GEMM.
// Grid: B * G * (HW/32);  block = 128 (4 waves).
//  - async-copy the group's 32x288 weight panel into LDS (overlapped with
//    the gather phase), retire with s_wait_asynccnt + barrier
//  - build sampled B-matrix S2 in LDS as K-pairs so fragments are ds_load_b64
//  - 2x2 grid of 16x16 fp32 WMMA tiles -> 32 out-ch x 32 pixels
// ---------------------------------------------------------------------------
__global__ __launch_bounds__(128) void deform_gemm_wmma(
    const float* __restrict__ x,        // [B][C][H][W]
    const float* __restrict__ offbuf,   // [B][144][H][W]
    const float* __restrict__ dw,       // [C][CG][3][3] -> row o: 288 floats
    const float* __restrict__ db,       // [C]
    float* __restrict__ out)            // [B][C][H][W]
{
    __shared__ float S2[(KTOT / 2) * S2S];  // 144*96 words = 55296 B
    __shared__ float Aw[CGC * AWS];         // 32*290  words = 37120 B

    const int bid   = blockIdx.x;
    const int pt    = bid & 127;        // pixel tile (32 px each)
    const int g     = (bid >> 7) & 7;
    const int b     = bid >> 10;
    const int tile0 = pt * 32;

    // ---- Phase 0: async-stage this group's weight panel into LDS ---------
    {
        const float* wsrc = dw + (size_t)g * CGC * KTOT;          // 9216 floats
        const unsigned awbase = (unsigned)(size_t)(void*)&Aw[0];  // LDS offset
        for (int i = threadIdx.x; i < CGC * KTOT; i += 128) {
            const int oo  = i / KTOT;
            const int kk2 = i - oo * KTOT;
            const unsigned ldst = awbase + (unsigned)((oo * AWS + kk2) * 4);
            const unsigned voff = (unsigned)(i * 4);
            asm volatile("global_load_async_to_lds_b32 %0, %1, %2"
                         :
                         : "v"(ldst), "v"(voff), "s"(wsrc)
                         : "memory");
        }
    }

    // ---- Phase 1: bilinear sampling into LDS (overlaps async copies) ------
    {
        const int t    = threadIdx.x;
        const int p    = t & 31;            // pixel within tile
        const int cblk = t >> 5;            // channel block: 8 channels each
        const int pix  = tile0 + p;
        const int hh   = pix >> 6;
        const int ww   = pix & 63;
        const float* xg = x + (((size_t)b * CC + g * CGC) * HW);

        for (int kk = 0; kk < KKK; ++kk) {
            const int ki  = kk / 3, kj = kk % 3;
            const int chy = (g * KKK + kk) * 2;
            const float oy = offbuf[((size_t)b * OFFC + chy) * HW + pix];
            const float ox = offbuf[((size_t)b * OFFC + chy + 1) * HW + pix];

            const float py  = (float)(hh - 1 + ki) + oy;
            const float px  = (float)(ww - 1 + kj) + ox;
            const float y0f = floorf(py), x0f = floorf(px);
            const float wy1 = py - y0f,  wx1 = px - x0f;
            const float wy0 = 1.0f - wy1, wx0 = 1.0f - wx1;
            const int iy0 = (int)y0f, ix0 = (int)x0f;
            const int iy1 = iy0 + 1,  ix1 = ix0 + 1;

            const float vy0 = (iy0 >= 0 && iy0 < HH) ? 1.0f : 0.0f;
            const float vy1 = (iy1 >= 0 && iy1 < HH) ? 1.0f : 0.0f;
            const float vx0 = (ix0 >= 0 && ix0 < WW) ? 1.0f : 0.0f;
            const float vx1 = (ix1 >= 0 && ix1 < WW) ? 1.0f : 0.0f;

            const float w00 = wy0 * wx0 * vy0 * vx0;
            const float w01 = wy0 * wx1 * vy0 * vx1;
            const float w10 = wy1 * wx0 * vy1 * vx0;
            const float w11 = wy1 * wx1 * vy1 * vx1;

            const int y0c = min(max(iy0, 0), HH - 1);
            const int y1c = min(max(iy1, 0), HH - 1);
            const int x0c = min(max(ix0, 0), WW - 1);
            const int x1c = min(max(ix1, 0), WW - 1);
            const int i00 = y0c * WW + x0c, i01 = y0c * WW + x1c;
            const int i10 = y1c * WW + x0c, i11 = y1c * WW + x1c;

#pragma unroll
            for (int cc = 0; cc < 8; ++cc) {
                const int c  = cblk * 8 + cc;
                const float* xc = xg + (size_t)c * HW;
                const float v = w00 * xc[i00] + w01 * xc[i01] +
                                w10 * xc[i10] + w11 * xc[i11];
                const int k = c * KKK + kk;
                // K-pair interleaved layout: one ds_load_b64 per B fragment
                S2[(k >> 1) * S2S + p * 2 + (k & 1)] = v;
            }
        }
    }

    asm volatile("s_wait_asynccnt 0" ::: "memory");   // weight panel resident
    __syncthreads();

    // ---- Phase 2: fp32 WMMA group GEMM (A and B both from LDS) ------------
    const int lane = threadIdx.x & 31;
    const int wv   = threadIdx.x >> 5;
    const int mi   = wv >> 1;              // M tile (0..1) -> 16 out-channels
    const int ni   = wv & 1;               // N tile (0..1) -> 16 pixels
    const int row  = lane & 15;
    const int kh2  = (lane < 16) ? 0 : 2;

    const int arow = (mi * 16 + row) * AWS;
    const int bcol = (ni * 16 + row) * 2;

    v8f acc = {};
#pragma unroll 4
    for (int k0 = 0; k0 < KTOT; k0 += 4) {
        v2f a  = *(const v2f*)&Aw[arow + k0 + kh2];
        v2f bm = *(const v2f*)&S2[((k0 + kh2) >> 1) * S2S + bcol];
        acc = __builtin_amdgcn_wmma_f32_16x16x4_f32(
            false, a, false, bm, (short)0, acc, false, false);
    }

    const int pcol = tile0 + ni * 16 + row;
    const int mh   = (lane < 16) ? 0 : 8;
#pragma unroll
    for (int v = 0; v < 8; ++v) {
        const int co = g * CGC + mi * 16 + v + mh;
        out[((size_t)b * CC + co) * HW + pcol] = acc[v] + db[co];
    }
}

// ---------------------------------------------------------------------------
extern "C" void kernel_launch(void* const* d_in, const int* in_sizes, int n_in,
                              void* d_out, int out_size, void* d_ws, size_t ws_size,
                              hipStream_t stream) {
    const float* x        = (const float*)d_in[0];  // [4][256][64][64]
    const float* offset_w = (const float*)d_in[1];  // [144][256][3][3]
    const float* offset_b = (const float*)d_in[2];  // [144]
    const float* deform_w = (const float*)d_in[3];  // [256][32][3][3]
    const float* deform_b = (const float*)d_in[4];  // [256]
    float*       out      = (float*)d_out;          // [4][256][64][64]

    float* offbuf = (float*)d_ws;                        // 4*144*4096 = 9.44 MB
    float* wT     = offbuf + (size_t)BB * OFFC * HW;     // 9*144*256  = 1.33 MB

    // Repack offset conv weights: 144*256*9 = 331776 elements
    repack_offw<<<(OFFC * CC * KKK + 255) / 256, 256, 0, stream>>>(offset_w, wT);

    // Kernel A: B * 9 m-tiles * H rows = 2304 blocks
    offset_conv_wmma<<<BB * 9 * HH, 128, 0, stream>>>(x, wT, offset_b, offbuf);

    // Kernel B: B * G * (HW/32) = 4096 blocks
    deform_gemm_wmma<<<BB * GG * (HW / 32), 128, 0, stream>>>(
        x, offbuf, deform_w, deform_b, out);
}